// LightGCNCover_61632780698015
// MI455X (gfx1250) — compile-verified
//
#include <hip/hip_runtime.h>

typedef __attribute__((ext_vector_type(2))) float v2f;
typedef __attribute__((ext_vector_type(8))) float v8f;

#define EMBED 64
#define COVER 512
#define N_LAYERS 3

// ---------------------------------------------------------------------------
// Hardware fp32 atomic add (global_atomic_add_f32), avoids CAS loop.
// ---------------------------------------------------------------------------
__device__ __forceinline__ void atomic_add_f32(float* p, float v) {
    __hip_atomic_fetch_add(p, v, __ATOMIC_RELAXED, __HIP_MEMORY_SCOPE_AGENT);
}

// ---------------------------------------------------------------------------
// Kernel 1: x[0:n_users] = user_emb ; acc[0:n_users] = user_emb   (float4)
// ---------------------------------------------------------------------------
__global__ void copy_users_kernel(const float4* __restrict__ user_emb,
                                  float4* __restrict__ x,
                                  float4* __restrict__ acc, int n4) {
    int i = blockIdx.x * blockDim.x + threadIdx.x;
    if (i < n4) {
        float4 v = user_emb[i];
        x[i] = v;
        acc[i] = v;
    }
}

// ---------------------------------------------------------------------------
// Kernel 2: cover GEMM via V_WMMA_F32_16X16X4_F32.
//   proj = item_cover[50000,512] @ cover_W.T   (cover_W is [64,512] row-major)
//   x[n_users + m] = item_emb[m] + has_cover[m] * proj[m] ; acc likewise.
// One wave handles one 16-row M tile and all four 16-col N tiles, so
// item_cover is streamed exactly once from HBM.
// A-layout (32-bit A 16x4): lane<16 -> K=k,k+1 ; lane>=16 -> K=k+2,k+3.
// B-layout mirrors A; C/D: c[g] = C[g + 8*(lane>>4)][lane&15].
// ---------------------------------------------------------------------------
__global__ void cover_gemm_wmma(const float* __restrict__ item_cover,
                                const float* __restrict__ cover_W,
                                const float* __restrict__ item_emb,
                                const float* __restrict__ has_cover,
                                float* __restrict__ x,
                                float* __restrict__ acc,
                                int n_users, int n_items) {
    const int lane = threadIdx.x & 31;
    const int wid  = threadIdx.x >> 5;
    const int tm   = blockIdx.x * (blockDim.x >> 5) + wid;   // M tile id
    const int n_mtiles = n_items >> 4;
    if (tm >= n_mtiles) return;                              // wave-uniform

    const int half = lane >> 4;   // 0/1 : which K pair
    const int l16  = lane & 15;

    const float* __restrict__ Arow = item_cover + (size_t)(tm * 16 + l16) * COVER;
    const float* __restrict__ B0 = cover_W + (size_t)( 0 + l16) * COVER;
    const float* __restrict__ B1 = cover_W + (size_t)(16 + l16) * COVER;
    const float* __restrict__ B2 = cover_W + (size_t)(32 + l16) * COVER;
    const float* __restrict__ B3 = cover_W + (size_t)(48 + l16) * COVER;

    v8f c0 = {}, c1 = {}, c2 = {}, c3 = {};
    #pragma unroll 4
    for (int k = 0; k < COVER; k += 4) {
        const int ka = k + half * 2;                         // 8B aligned
        v2f a  = *(const v2f*)(Arow + ka);
        v2f b0 = *(const v2f*)(B0 + ka);
        v2f b1 = *(const v2f*)(B1 + ka);
        v2f b2 = *(const v2f*)(B2 + ka);
        v2f b3 = *(const v2f*)(B3 + ka);
        c0 = __builtin_amdgcn_wmma_f32_16x16x4_f32(false, a, false, b0, (short)0, c0, false, false);
        c1 = __builtin_amdgcn_wmma_f32_16x16x4_f32(false, a, false, b1, (short)0, c1, false, false);
        c2 = __builtin_amdgcn_wmma_f32_16x16x4_f32(false, a, false, b2, (short)0, c2, false, false);
        c3 = __builtin_amdgcn_wmma_f32_16x16x4_f32(false, a, false, b3, (short)0, c3, false, false);
    }

    #pragma unroll
    for (int g = 0; g < 8; ++g) {
        const int m  = tm * 16 + g + 8 * half;               // item row
        const float hc = has_cover[m];
        const size_t eb = (size_t)m * EMBED;
        const size_t ob = (size_t)(n_users + m) * EMBED;
        float v0 = c0[g] * hc + item_emb[eb +  0 + l16];
        float v1 = c1[g] * hc + item_emb[eb + 16 + l16];
        float v2 = c2[g] * hc + item_emb[eb + 32 + l16];
        float v3 = c3[g] * hc + item_emb[eb + 48 + l16];
        x[ob +  0 + l16] = v0;  acc[ob +  0 + l16] = v0;
        x[ob + 16 + l16] = v1;  acc[ob + 16 + l16] = v1;
        x[ob + 32 + l16] = v2;  acc[ob + 32 + l16] = v2;
        x[ob + 48 + l16] = v3;  acc[ob + 48 + l16] = v3;
    }
}

// ---------------------------------------------------------------------------
// Kernel 3: zero a buffer (float4)
// ---------------------------------------------------------------------------
__global__ void zero_kernel(float4* __restrict__ p, int n4) {
    int i = blockIdx.x * blockDim.x + threadIdx.x;
    if (i < n4) p[i] = make_float4(0.f, 0.f, 0.f, 0.f);
}

// ---------------------------------------------------------------------------
// Kernel 4: COO SpMM edge scatter.  16 lanes per edge, one float4 per lane:
//   y[rows[e]] += vals[e] * x[cols[e]]      (64 floats per edge)
// Half-wave loads 256 contiguous bytes of x[col] -> fully coalesced gather.
// ---------------------------------------------------------------------------
__global__ void spmm_edge_kernel(const float* __restrict__ vals,
                                 const int* __restrict__ rows,
                                 const int* __restrict__ cols,
                                 const float* __restrict__ x,
                                 float* __restrict__ y,
                                 int n_edges) {
    long long idx = (long long)blockIdx.x * blockDim.x + threadIdx.x;
    int e = (int)(idx >> 4);
    int c = (int)(idx & 15);
    if (e >= n_edges) return;

    const float v  = vals[e];
    const int  col = cols[e];
    const int  row = rows[e];

    const float4 xs = *((const float4*)(x + (size_t)col * EMBED) + c);
    float* dst = (float*)((float4*)(y + (size_t)row * EMBED) + c);
    atomic_add_f32(dst + 0, v * xs.x);
    atomic_add_f32(dst + 1, v * xs.y);
    atomic_add_f32(dst + 2, v * xs.z);
    atomic_add_f32(dst + 3, v * xs.w);
}

// ---------------------------------------------------------------------------
// Kernel 5: acc = (acc + y) * scale   (float4); final layer uses scale=1/4.
// ---------------------------------------------------------------------------
__global__ void acc_kernel(float4* __restrict__ acc,
                           const float4* __restrict__ y,
                           float scale, int n4) {
    int i = blockIdx.x * blockDim.x + threadIdx.x;
    if (i < n4) {
        float4 a = acc[i], b = y[i];
        a.x = (a.x + b.x) * scale;
        a.y = (a.y + b.y) * scale;
        a.z = (a.z + b.z) * scale;
        a.w = (a.w + b.w) * scale;
        acc[i] = a;
    }
}

// ---------------------------------------------------------------------------
// Inputs: 0 user_emb, 1 item_emb, 2 cover_W, 3 item_cover, 4 has_cover,
//         5 adj_vals, 6 adj_rows, 7 adj_cols, 8 n_layers (device scalar; =3)
// Output: concat(user rows, item rows) of acc/(n_layers+1) = [n_nodes,64].
// Workspace: two ping-pong node buffers (2 * n_nodes * 64 floats).
// acc is stored directly in d_out.
// ---------------------------------------------------------------------------
extern "C" void kernel_launch(void* const* d_in, const int* in_sizes, int n_in,
                              void* d_out, int out_size, void* d_ws, size_t ws_size,
                              hipStream_t stream) {
    const float* user_emb   = (const float*)d_in[0];
    const float* item_emb   = (const float*)d_in[1];
    const float* cover_W    = (const float*)d_in[2];
    const float* item_cover = (const float*)d_in[3];
    const float* has_cover  = (const float*)d_in[4];
    const float* adj_vals   = (const float*)d_in[5];
    const int*   adj_rows   = (const int*)d_in[6];
    const int*   adj_cols   = (const int*)d_in[7];

    const int n_users = in_sizes[0] / EMBED;
    const int n_items = in_sizes[1] / EMBED;
    const int n_edges = in_sizes[5];
    const int n_nodes = n_users + n_items;

    float* acc  = (float*)d_out;                       // [n_nodes, 64]
    float* bufA = (float*)d_ws;                        // [n_nodes, 64]
    float* bufB = bufA + (size_t)n_nodes * EMBED;      // [n_nodes, 64]

    // --- build x = concat(user_emb, item_emb + has_cover * cover_proj); acc = x
    {
        int n4 = n_users * (EMBED / 4);
        copy_users_kernel<<<(n4 + 255) / 256, 256, 0, stream>>>(
            (const float4*)user_emb, (float4*)bufA, (float4*)acc, n4);

        const int waves_per_block = 4;                 // 128 threads
        int n_mtiles = n_items / 16;
        int grid = (n_mtiles + waves_per_block - 1) / waves_per_block;
        cover_gemm_wmma<<<grid, waves_per_block * 32, 0, stream>>>(
            item_cover, cover_W, item_emb, has_cover, bufA, acc, n_users, n_items);
    }

    // --- 3 propagation layers: y = A @ x ; acc += y ; x <- y (ping-pong)
    const int nodes4 = n_nodes * (EMBED / 4);
    const long long edge_threads = (long long)n_edges * 16;
    const int edge_blocks = (int)((edge_threads + 255) / 256);

    float* x = bufA;
    float* y = bufB;
    for (int layer = 0; layer < N_LAYERS; ++layer) {
        zero_kernel<<<(nodes4 + 255) / 256, 256, 0, stream>>>((float4*)y, nodes4);
        spmm_edge_kernel<<<edge_blocks, 256, 0, stream>>>(
            adj_vals, adj_rows, adj_cols, x, y, n_edges);
        const float scale = (layer == N_LAYERS - 1)
                          ? 1.0f / (float)(N_LAYERS + 1) : 1.0f;
        acc_kernel<<<(nodes4 + 255) / 256, 256, 0, stream>>>(
            (float4*)acc, (const float4*)y, scale, nodes4);
        float* t = x; x = y; y = t;
    }
}